// MultiHeadGATConv_48808008352177
// MI455X (gfx1250) — compile-verified
//
#include <hip/hip_runtime.h>
#include <hip/hip_bf16.h>

typedef __attribute__((ext_vector_type(2))) float v2f;
typedef __attribute__((ext_vector_type(8))) float v8f;

#define H_HEADS 4
#define F_DIM   128                 // F_IN == F_OUT == 128
#define HF      (H_HEADS * F_DIM)   // 512
#define PSTR    160                 // floats per K-pair row: 160 % 64 == 32 -> conflict-free b64
#define PSTR2   (PSTR / 2)          // v2f elements per K-pair row

// ---------------------------------------------------------------------------
// Kernel 1: xW[n][h*128+f] = sum_k x[n][k] * W[h][k][f]
// Block = 4 waves, owns one 64-col group; stages its [128][64] W slice in LDS
// with K-pairs interleaved so each WMMA B fragment is one ds_load_b64.
// Grid-strides over 16-row tiles. WMMA f32 16x16x4, K steps of 4.
// ---------------------------------------------------------------------------
__global__ __launch_bounds__(128) void gat_xw_gemm(const float* __restrict__ x,
                                                   const float* __restrict__ W,
                                                   float* __restrict__ xW,
                                                   int rowTiles) {
    extern __shared__ float ldsW[];          // [64 k-pairs][PSTR]
    const int tid  = threadIdx.x;
    const int lane = tid & 31;
    const int wIn  = tid >> 5;
    const int colGrp    = blockIdx.x & 7;    // 8 groups of 64 over 512 cols
    const int grpBlk    = blockIdx.x >> 3;
    const int grpBlocks = gridDim.x  >> 3;
    const int wavesPerGrp = grpBlocks * (blockDim.x >> 5);
    const int waveInGrp   = grpBlk * (blockDim.x >> 5) + wIn;

    // stage B slice: head = colGrp>>1, cols [(colGrp&1)*64, +64), pair-interleaved
    const float* Wh = W + (size_t)(colGrp >> 1) * (F_DIM * F_DIM) + (colGrp & 1) * 64;
    for (int idx = tid; idx < F_DIM * 64; idx += blockDim.x) {
        const int k = idx >> 6, c = idx & 63;
        ldsW[(k >> 1) * PSTR + (c << 1) + (k & 1)] = Wh[(size_t)k * F_DIM + c];
    }
    __syncthreads();

    const int half = lane >> 4;              // selects K pair within WMMA fragment
    const int mrow = lane & 15;
    const int colBase = colGrp * 64;
    const v2f* ldsP = (const v2f*)ldsW;

    for (int rt = waveInGrp; rt < rowTiles; rt += wavesPerGrp) {  // wave-uniform
        const int row0 = rt * 16;
        const v2f* xrow = (const v2f*)(x + (size_t)(row0 + mrow) * F_DIM + 2 * half);

        v8f acc[4];
#pragma unroll
        for (int t = 0; t < 4; ++t) acc[t] = (v8f){0, 0, 0, 0, 0, 0, 0, 0};

        for (int k0 = 0; k0 < F_DIM; k0 += 4) {
            const v2f a = xrow[k0 >> 1];     // A 16x4 fragment (b64 load)
            const int kp = (k0 >> 1) + half; // K-pair row index
#pragma unroll
            for (int t = 0; t < 4; ++t) {
                const v2f b = ldsP[kp * PSTR2 + t * 16 + mrow];  // one ds_load_b64
                acc[t] = __builtin_amdgcn_wmma_f32_16x16x4_f32(
                    false, a, false, b, (short)0, acc[t], false, false);
            }
        }

#pragma unroll
        for (int t = 0; t < 4; ++t) {
            const int j = colBase + t * 16 + mrow;
#pragma unroll
            for (int r = 0; r < 8; ++r) {
                const int row = row0 + r + half * 8;   // C/D layout: VGPR r -> M=r / M=r+8
                xW[(size_t)row * HF + j] = acc[t][r];
            }
        }
    }
}

// ---------------------------------------------------------------------------
// Kernel 2: per (node, head) attention coefficients + state init.
// ---------------------------------------------------------------------------
__global__ __launch_bounds__(256) void gat_alpha(const float* __restrict__ xW,
                          const float* __restrict__ att_src,
                          const float* __restrict__ att_dst,
                          float* __restrict__ alpha_s,
                          float* __restrict__ alpha_d,
                          unsigned* __restrict__ smax,
                          float* __restrict__ denom,
                          float* __restrict__ agg,
                          int NH) {
    const int wave = blockIdx.x * (blockDim.x >> 5) + (threadIdx.x >> 5);
    const int lane = threadIdx.x & 31;
    if (wave >= NH) return;             // wave-uniform
    const int n = wave >> 2;
    const int h = wave & 3;

    const float* row = xW + (size_t)n * HF + h * F_DIM;
    const float* vs  = att_src + h * F_DIM;
    const float* vd  = att_dst + h * F_DIM;
    float* arow      = agg + (size_t)n * HF + h * F_DIM;

    float ss = 0.f, sd = 0.f;
#pragma unroll
    for (int o = 0; o < F_DIM; o += 32) {
        const float v = row[lane + o];
        ss += v * vs[lane + o];
        sd += v * vd[lane + o];
        arow[lane + o] = 0.f;           // zero aggregation buffer
    }
#pragma unroll
    for (int off = 16; off > 0; off >>= 1) {
        ss += __shfl_down(ss, off, 32);
        sd += __shfl_down(sd, off, 32);
    }
    if (lane == 0) {
        alpha_s[wave] = ss;
        alpha_d[wave] = sd;
        smax[wave]  = 0u;               // lowest value under monotone uint encoding
        denom[wave] = 0.f;
    }
}

// monotone float<->uint map (order preserving) for atomicMax on floats
__device__ __forceinline__ unsigned fmax_key(float f) {
    unsigned u = __float_as_uint(f);
    return (u & 0x80000000u) ? ~u : (u | 0x80000000u);
}
__device__ __forceinline__ float fmax_unkey(unsigned u) {
    return (u & 0x80000000u) ? __uint_as_float(u & 0x7FFFFFFFu)
                             : __uint_as_float(~u);
}

// ---------------------------------------------------------------------------
// Kernel 3: edge logits (leaky relu) + segment max over dst (all 4 heads).
// ---------------------------------------------------------------------------
__global__ __launch_bounds__(256) void gat_edge_max(const int* __restrict__ ei, int E, int ET,
                             const float* __restrict__ alpha_s,
                             const float* __restrict__ alpha_d,
                             float* __restrict__ elog,
                             unsigned* __restrict__ smax) {
    const int e = blockIdx.x * blockDim.x + threadIdx.x;
    if (e >= ET) return;
    int s, d;
    if (e < E) { s = ei[e]; d = ei[E + e]; } else { s = d = e - E; }
#pragma unroll
    for (int h = 0; h < H_HEADS; ++h) {
        float l = alpha_s[s * H_HEADS + h] + alpha_d[d * H_HEADS + h];
        l = l > 0.f ? l : 0.2f * l;     // LeakyReLU, slope 0.2
        elog[(size_t)e * H_HEADS + h] = l;
        atomicMax(&smax[d * H_HEADS + h], fmax_key(l));
    }
}

// ---------------------------------------------------------------------------
// Kernel 4: e = exp(logit - segmax[dst]); store in place; denom[dst] += e
// ---------------------------------------------------------------------------
__global__ __launch_bounds__(256) void gat_edge_exp(const int* __restrict__ ei, int E, int ET,
                             const unsigned* __restrict__ smax,
                             float* __restrict__ elog,
                             float* __restrict__ denom) {
    const int e = blockIdx.x * blockDim.x + threadIdx.x;
    if (e >= ET) return;
    int d;
    if (e < E) { d = ei[E + e]; } else { d = e - E; }
#pragma unroll
    for (int h = 0; h < H_HEADS; ++h) {
        const float m  = fmax_unkey(smax[d * H_HEADS + h]);
        const float ex = __expf(elog[(size_t)e * H_HEADS + h] - m);
        elog[(size_t)e * H_HEADS + h] = ex;
        unsafeAtomicAdd(&denom[d * H_HEADS + h], ex);
    }
}

// ---------------------------------------------------------------------------
// Kernel 5: agg[dst][h*128+f] += (e/denom[dst][h]) * xW[src][h*128+f]
// One wave per (edge, head). xW and agg (102 MB each) sit in the 192 MB L2,
// so the random gather + f32 atomics mostly stay on-chip.
// ---------------------------------------------------------------------------
__global__ __launch_bounds__(256) void gat_aggregate(const int* __restrict__ ei, int E, int P,
                              const float* __restrict__ elog,
                              const float* __restrict__ denom,
                              const float* __restrict__ xW,
                              float* __restrict__ agg) {
    const int wave = blockIdx.x * (blockDim.x >> 5) + (threadIdx.x >> 5);
    const int lane = threadIdx.x & 31;
    if (wave >= P) return;              // wave-uniform
    const int e = wave >> 2;
    const int h = wave & 3;
    int s, d;
    if (e < E) { s = ei[e]; d = ei[E + e]; } else { s = d = e - E; }

    const float a = elog[(size_t)e * H_HEADS + h] / denom[d * H_HEADS + h];
    const float* src = xW + (size_t)s * HF + h * F_DIM;
    float* dst       = agg + (size_t)d * HF + h * F_DIM;
#pragma unroll
    for (int o = 0; o < F_DIM; o += 32)
        unsafeAtomicAdd(&dst[lane + o], a * src[lane + o]);
}

// ---------------------------------------------------------------------------
// Kernel 6: out = relu((agg + bias) @ W_out + b_out), K = 512, WMMA f32.
// Block stages its full-K [512][64] W_out slice pair-interleaved in LDS
// (160 KB of the 320 KB WGP LDS), then grid-strides over row tiles.
// ---------------------------------------------------------------------------
__global__ __launch_bounds__(128) void gat_out_gemm(const float* __restrict__ agg,
                             const float* __restrict__ bias,     // 512 floats
                             const float* __restrict__ W_out,    // [512][128]
                             const float* __restrict__ b_out,    // 128 floats
                             float* __restrict__ out,
                             int rowTiles) {
    extern __shared__ float ldsW[];          // [256 k-pairs][PSTR]
    const int tid  = threadIdx.x;
    const int lane = tid & 31;
    const int wIn  = tid >> 5;
    const int colGrp    = blockIdx.x & 1;    // 2 groups of 64 over 128 cols
    const int grpBlk    = blockIdx.x >> 1;
    const int grpBlocks = gridDim.x  >> 1;
    const int wavesPerGrp = grpBlocks * (blockDim.x >> 5);
    const int waveInGrp   = grpBlk * (blockDim.x >> 5) + wIn;

    const int colBase = colGrp * 64;
    for (int idx = tid; idx < HF * 64; idx += blockDim.x) {
        const int k = idx >> 6, c = idx & 63;
        ldsW[(k >> 1) * PSTR + (c << 1) + (k & 1)] = W_out[(size_t)k * F_DIM + colBase + c];
    }
    __syncthreads();

    const int half = lane >> 4;
    const int mrow = lane & 15;
    const v2f* ldsP = (const v2f*)ldsW;

    for (int rt = waveInGrp; rt < rowTiles; rt += wavesPerGrp) {  // wave-uniform
        const int row0 = rt * 16;
        const float* arow = agg  + (size_t)(row0 + mrow) * HF + 2 * half;
        const float* brow = bias + 2 * half;

        v8f acc[4];
#pragma unroll
        for (int t = 0; t < 4; ++t) acc[t] = (v8f){0, 0, 0, 0, 0, 0, 0, 0};

        for (int k0 = 0; k0 < HF; k0 += 4) {
            v2f a;
            a.x = arow[k0]     + brow[k0];       // fold per-head bias into h_cat
            a.y = arow[k0 + 1] + brow[k0 + 1];
            const int kp = (k0 >> 1) + half;     // K-pair row index
#pragma unroll
            for (int t = 0; t < 4; ++t) {
                const v2f b = ldsP[kp * PSTR2 + t * 16 + mrow];  // one ds_load_b64
                acc[t] = __builtin_amdgcn_wmma_f32_16x16x4_f32(
                    false, a, false, b, (short)0, acc[t], false, false);
            }
        }

#pragma unroll
        for (int t = 0; t < 4; ++t) {
            const int col = colBase + t * 16 + mrow;
            const float bo = b_out[col];
#pragma unroll
            for (int r = 0; r < 8; ++r) {
                const int row = row0 + r + half * 8;
                const float v = acc[t][r] + bo;
                out[(size_t)row * F_DIM + col] = v > 0.f ? v : 0.f;
            }
        }
    }
}

// ---------------------------------------------------------------------------
extern "C" void kernel_launch(void* const* d_in, const int* in_sizes, int n_in,
                              void* d_out, int out_size, void* d_ws, size_t ws_size,
                              hipStream_t stream) {
    const float* x       = (const float*)d_in[0];
    const int*   ei      = (const int*)d_in[1];   // [2][E] (src row, dst row)
    const float* W       = (const float*)d_in[2]; // [H][128][128]
    const float* att_src = (const float*)d_in[3]; // [H][128]
    const float* att_dst = (const float*)d_in[4];
    const float* bias    = (const float*)d_in[5]; // [H][128] -> flat 512
    const float* W_out   = (const float*)d_in[6]; // [512][128]
    const float* b_out   = (const float*)d_in[7]; // [128]
    float* out = (float*)d_out;

    const int N  = in_sizes[0] / F_DIM;
    const int E  = in_sizes[1] / 2;
    const int ET = E + N;                 // edges + self loops
    const int NH = N * H_HEADS;

    // workspace carve-up (floats)
    float*    ws      = (float*)d_ws;
    float*    xW      = ws;                               // N*512
    float*    agg     = xW  + (size_t)N * HF;             // N*512
    float*    alpha_s = agg + (size_t)N * HF;             // N*4
    float*    alpha_d = alpha_s + (size_t)NH;             // N*4
    unsigned* smax    = (unsigned*)(alpha_d + (size_t)NH);// N*4
    float*    denom   = (float*)smax + (size_t)NH;        // N*4
    float*    elog    = denom + (size_t)NH;               // ET*4

    const int rowTiles = (N + 15) / 16;   // N = 50000 -> 3125 exact

    // 1) xW GEMM: 8 col-groups x 128 blocks, 40 KB LDS each
    gat_xw_gemm<<<8 * 128, 128, (F_DIM / 2) * PSTR * sizeof(float), stream>>>(x, W, xW, rowTiles);
    // 2) alpha + init: NH waves, 8 waves/block
    gat_alpha<<<(NH + 7) / 8, 256, 0, stream>>>(xW, att_src, att_dst,
                                                alpha_s, alpha_d, smax, denom,
                                                agg, NH);
    // 3) segment max
    gat_edge_max<<<(ET + 255) / 256, 256, 0, stream>>>(ei, E, ET,
                                                       alpha_s, alpha_d, elog, smax);
    // 4) exp + denom
    gat_edge_exp<<<(ET + 255) / 256, 256, 0, stream>>>(ei, E, ET, smax, elog, denom);
    // 5) weighted scatter-aggregate: ET*4 waves, 8 waves/block
    {
        int P = ET * H_HEADS;
        gat_aggregate<<<(P + 7) / 8, 256, 0, stream>>>(ei, E, P, elog, denom, xW, agg);
    }
    // 6) output GEMM + relu: 2 col-groups x 128 blocks, 160 KB LDS each
    gat_out_gemm<<<2 * 128, 128, (HF / 2) * PSTR * sizeof(float), stream>>>(agg, bias, W_out, b_out,
                                                                            out, rowTiles);
}